// PCDONN_88450556494244
// MI455X (gfx1250) — compile-verified
//
#include <hip/hip_runtime.h>
#include <hip/hip_bf16.h>
#include <math.h>

// ---------------------------------------------------------------------------
// PCDONN forward for MI455X (gfx1250), compile-only target.
// fft2/ifft2 are realized as 64x64x64 complex matmuls on V_WMMA_F32_16X16X4_F32
// so the whole per-field pipeline (3 layers, 6 transforms) stays in LDS.
// ---------------------------------------------------------------------------

#define RESO   64
#define NPIX   4096          // 64*64
#define MSCR   50
#define NBATCH 128
#define NLAY   3
#define NCLS   10

typedef __attribute__((ext_vector_type(2))) float v2f;
typedef __attribute__((ext_vector_type(8))) float v8f;

// ---------------------------------------------------------------------------
// Complex 64x64x64 matmul D = X * Y on the WMMA pipe.
//   X,Y,D: row-major 64x64 float re/im planes in LDS.
//   sigX/sigY: +-1.0f compile-time-folded signs on the imaginary parts
//              (implements conj(F) for the inverse transform).
//   scale: applied at writeback (1/N^2 for ifft).
// 8 waves per block; 16 output tiles of 16x16 -> 2 tiles per wave.
// ---------------------------------------------------------------------------
__device__ __forceinline__ void cmm64(const float* __restrict__ Xr,
                                      const float* __restrict__ Xi, float sigX,
                                      const float* __restrict__ Yr,
                                      const float* __restrict__ Yi, float sigY,
                                      float* __restrict__ Dr,
                                      float* __restrict__ Di, float scale) {
    const int lane = threadIdx.x & 31;
    const int wave = threadIdx.x >> 5;   // 0..7
    const int half = lane >> 4;          // 0/1 (K split per ISA layout)
    const int l15  = lane & 15;

    for (int t = wave; t < 16; t += 8) {             // uniform: 2 iters per wave
        const int ti = t >> 2;                       // tile row (M/16)
        const int tj = t & 3;                        // tile col (N/16)
        v8f accr = {};
        v8f acci = {};
        const float* arow = Xr + (ti * 16 + l15) * RESO + 2 * half;
        const float* brow = Xi + (ti * 16 + l15) * RESO + 2 * half;
        const int bcol = tj * 16 + l15;

        for (int k = 0; k < 16; ++k) {
            const int kc = k * 4;
            // A fragment: A[M=l15][K = kc + 2*half + {0,1}] (contiguous pair)
            v2f axr, axi, byr, byi;
            axr.x = arow[kc + 0];
            axr.y = arow[kc + 1];
            axi.x = brow[kc + 0] * sigX;
            axi.y = brow[kc + 1] * sigX;
            // B fragment: B[K = kc + 2*half + {0,1}][N = bcol]
            const int br = kc + 2 * half;
            byr.x = Yr[(br + 0) * RESO + bcol];
            byr.y = Yr[(br + 1) * RESO + bcol];
            byi.x = Yi[(br + 0) * RESO + bcol] * sigY;
            byi.y = Yi[(br + 1) * RESO + bcol] * sigY;
            v2f naxi = -axi;
            // Dr += Xr*Yr - Xi*Yi ; Di += Xr*Yi + Xi*Yr
            accr = __builtin_amdgcn_wmma_f32_16x16x4_f32(false, axr, false, byr,
                                                         (short)0, accr, false, false);
            accr = __builtin_amdgcn_wmma_f32_16x16x4_f32(false, naxi, false, byi,
                                                         (short)0, accr, false, false);
            acci = __builtin_amdgcn_wmma_f32_16x16x4_f32(false, axr, false, byi,
                                                         (short)0, acci, false, false);
            acci = __builtin_amdgcn_wmma_f32_16x16x4_f32(false, axi, false, byr,
                                                         (short)0, acci, false, false);
        }
        // C/D layout: vgpr r -> M = 8*half + r, N = l15 (within tile)
        const int baseM = ti * 16 + half * 8;
        const int baseN = tj * 16 + l15;
        for (int r = 0; r < 8; ++r) {
            Dr[(baseM + r) * RESO + baseN] = accr[r] * scale;
            Di[(baseM + r) * RESO + baseN] = acci[r] * scale;
        }
    }
}

// ---------------------------------------------------------------------------
// Init: DFT matrix F, transfer function H (angle reduced mod 2pi in double),
// and per-layer phase screens exp(i*phi) -> consts[] in workspace.
// consts layout (floats): Fr[4096] Fi[4096] Hr[4096] Hi[4096]
//                         Er[3*4096] Ei[3*4096]
// ---------------------------------------------------------------------------
__global__ void donn_init(const float* __restrict__ phase_masks,
                          float* __restrict__ consts) {
    const int t = blockIdx.x * blockDim.x + threadIdx.x;
    if (t >= NPIX) return;
    const int row = t >> 6;
    const int col = t & 63;

    // F[j][k] = exp(-2*pi*i*j*k/64); integer mod keeps the angle exact/small.
    const float TWO_PI = 6.28318530717958647692f;
    const int jk = (row * col) & 63;
    const float fang = -TWO_PI * (float)jk * (1.0f / 64.0f);
    float fs, fc;
    sincosf(fang, &fs, &fc);
    consts[t]        = fc;   // Fr
    consts[NPIX + t] = fs;   // Fi

    // H = exp(i*K*D) * exp(-i*pi*lambda*D*(fx^2+fy^2)); K*D ~ 4.1e5 rad, so
    // build + range-reduce the angle in double before float sincos.
    const double DPI    = 3.141592653589793238462643;
    const double LAMB   = 5.32e-7;
    const double DIST   = 0.035;
    const double PITCH  = 1.0e-6;
    const double fx = (double)((col < 32) ? col : col - 64) / (64.0 * PITCH);
    const double fy = (double)((row < 32) ? row : row - 64) / (64.0 * PITCH);
    double ang = (2.0 * DPI / LAMB) * DIST - DPI * LAMB * DIST * (fx * fx + fy * fy);
    ang = fmod(ang, 2.0 * DPI);
    float hs, hc;
    sincosf((float)ang, &hs, &hc);
    consts[2 * NPIX + t] = hc;  // Hr
    consts[3 * NPIX + t] = hs;  // Hi

    // Per-layer exp(i*phi_l)
    for (int l = 0; l < NLAY; ++l) {
        float es, ec;
        sincosf(phase_masks[l * NPIX + t], &es, &ec);
        consts[4 * NPIX + l * NPIX + t]               = ec;  // Er
        consts[(4 + NLAY) * NPIX + l * NPIX + t]      = es;  // Ei
    }
}

__global__ void donn_zero(float* __restrict__ p, int n) {
    const int t = blockIdx.x * blockDim.x + threadIdx.x;
    if (t < n) p[t] = 0.0f;
}

// ---------------------------------------------------------------------------
// Main field kernel: one block per (m,b) field. Entire 3-layer optical stack
// runs from LDS; only inputs in, intensity-atomics out.
// Dynamic LDS: Ar Ai Br Bi Fr Fi (6 * 4096 floats = 96 KB).
// ---------------------------------------------------------------------------
__global__ __launch_bounds__(256) void donn_field(
    const float* __restrict__ x, const float* __restrict__ scr_re,
    const float* __restrict__ scr_im, const float* __restrict__ consts,
    float* __restrict__ inten) {
    extern __shared__ float lds[];
    float* Ar = lds;
    float* Ai = lds + NPIX;
    float* Br = lds + 2 * NPIX;
    float* Bi = lds + 3 * NPIX;
    float* Fr = lds + 4 * NPIX;
    float* Fi = lds + 5 * NPIX;

    const float* gFr = consts;
    const float* gFi = consts + NPIX;
    const float* gHr = consts + 2 * NPIX;
    const float* gHi = consts + 3 * NPIX;
    const float* gEr = consts + 4 * NPIX;
    const float* gEi = consts + (4 + NLAY) * NPIX;

    const int b = blockIdx.x % NBATCH;
    const int m = blockIdx.x / NBATCH;
    const int tid = threadIdx.x;

    const float* xb = x + b * NPIX;       // x[b,0,:,:] (real)
    const float* sr = scr_re + m * NPIX;
    const float* si = scr_im + m * NPIX;
    for (int i = tid; i < NPIX; i += 256) {
        const float xv = xb[i];
        Ar[i] = xv * sr[i];               // field = x * screen
        Ai[i] = xv * si[i];
        Fr[i] = gFr[i];                   // stage F into LDS
        Fi[i] = gFi[i];
    }
    __syncthreads();

    for (int l = 0; l < NLAY; ++l) {
        // field *= exp(i*phi_l)
        const float* er = gEr + l * NPIX;
        const float* ei = gEi + l * NPIX;
        for (int i = tid; i < NPIX; i += 256) {
            const float ar = Ar[i], ai = Ai[i];
            const float pr = er[i], pi = ei[i];
            Ar[i] = ar * pr - ai * pi;
            Ai[i] = ar * pi + ai * pr;
        }
        __syncthreads();
        // fft2: B = F*A ; A = B*F
        cmm64(Fr, Fi, 1.0f, Ar, Ai, 1.0f, Br, Bi, 1.0f);
        __syncthreads();
        cmm64(Br, Bi, 1.0f, Fr, Fi, 1.0f, Ar, Ai, 1.0f);
        __syncthreads();
        // spectrum *= H
        for (int i = tid; i < NPIX; i += 256) {
            const float ar = Ar[i], ai = Ai[i];
            const float hr = gHr[i], hi = gHi[i];
            Ar[i] = ar * hr - ai * hi;
            Ai[i] = ar * hi + ai * hr;
        }
        __syncthreads();
        // ifft2: B = conj(F)*A ; A = (B*conj(F)) / N^2
        cmm64(Fr, Fi, -1.0f, Ar, Ai, 1.0f, Br, Bi, 1.0f);
        __syncthreads();
        cmm64(Br, Bi, 1.0f, Fr, Fi, -1.0f, Ar, Ai, 1.0f / 4096.0f);
        __syncthreads();
    }

    // intensity accumulation: mean over M screens
    float* ib = inten + b * NPIX;
    const float invM = 1.0f / (float)MSCR;
    for (int i = tid; i < NPIX; i += 256) {
        const float v = (Ar[i] * Ar[i] + Ai[i] * Ai[i]) * invM;
        atomicAdd(&ib[i], v);
    }
}

// ---------------------------------------------------------------------------
// FC: one wave32 per (b,c) output, 4096-wide dot + shuffle reduction.
// ---------------------------------------------------------------------------
__global__ __launch_bounds__(256) void donn_fc(const float* __restrict__ inten,
                                               const float* __restrict__ w,
                                               const float* __restrict__ bias,
                                               float* __restrict__ out) {
    const int gw = (int)((blockIdx.x * blockDim.x + threadIdx.x) >> 5);
    const int lane = threadIdx.x & 31;
    if (gw >= NBATCH * NCLS) return;
    const int b = gw / NCLS;
    const int c = gw % NCLS;
    const float* ib = inten + b * NPIX;
    const float* wc = w + c * NPIX;
    float s = 0.0f;
    for (int k = lane; k < NPIX; k += 32) s += ib[k] * wc[k];
    for (int off = 16; off > 0; off >>= 1) s += __shfl_down(s, off, 32);
    if (lane == 0) out[b * NCLS + c] = s + bias[c];
}

extern "C" void kernel_launch(void* const* d_in, const int* in_sizes, int n_in,
                              void* d_out, int out_size, void* d_ws, size_t ws_size,
                              hipStream_t stream) {
    const float* x      = (const float*)d_in[0];  // [128,1,64,64]
    const float* scr_re = (const float*)d_in[1];  // [50,64,64]
    const float* scr_im = (const float*)d_in[2];  // [50,64,64]
    const float* pmasks = (const float*)d_in[3];  // [3,64,64]
    const float* fc_w   = (const float*)d_in[4];  // [10,4096]
    const float* fc_b   = (const float*)d_in[5];  // [10]
    float* out = (float*)d_out;                   // [128,10]

    float* ws     = (float*)d_ws;
    float* inten  = ws;                           // 128*4096 floats
    float* consts = ws + NBATCH * NPIX;           // (4 + 2*NLAY)*4096 floats

    donn_init<<<(NPIX + 255) / 256, 256, 0, stream>>>(pmasks, consts);
    donn_zero<<<(NBATCH * NPIX + 255) / 256, 256, 0, stream>>>(inten, NBATCH * NPIX);
    donn_field<<<MSCR * NBATCH, 256, 6 * NPIX * sizeof(float), stream>>>(
        x, scr_re, scr_im, consts, inten);
    donn_fc<<<(NBATCH * NCLS * 32 + 255) / 256, 256, 0, stream>>>(inten, fc_w, fc_b, out);
    (void)in_sizes; (void)n_in; (void)out_size; (void)ws_size;
}